// GNNmodel_72344429134240
// MI455X (gfx1250) — compile-verified
//
#include <hip/hip_runtime.h>
#include <math.h>

#define N_NODES 4096
#define N_EDGES 32768

typedef __attribute__((ext_vector_type(2))) float v2f;
typedef __attribute__((ext_vector_type(4))) float v4f;
typedef __attribute__((ext_vector_type(8))) float v8f;

#if __has_builtin(__builtin_amdgcn_tanhf)
#define TANHF(x) __builtin_amdgcn_tanhf(x)
#elif __has_builtin(__builtin_amdgcn_tanh_f32)
#define TANHF(x) __builtin_amdgcn_tanh_f32(x)
#else
#define TANHF(x) tanhf(x)
#endif

// ---------------------------------------------------------------------------
// One-wave fp32 GEMM layer on V_WMMA_F32_16X16X4_F32, fully static shapes.
// Inputs are zero-padded in memory (LDS activations and packed weights), so
// every fragment load is an unconditional 8-byte load: no exec-mask branches.
//   out[16][DOUTP] = act(in[16][DINP] @ Wp^T + bp)
// Wp is [DOUTP][DINP] zero-padded row-major, bp is [DOUTP] zero-padded.
// ISA 7.12.2: A 16x4 fp32 (lanes 0-15 K={0,1}, lanes 16-31 K={2,3});
// D 16x16 fp32 (lane<16 -> M=v, lane>=16 -> M=v+8).
// ---------------------------------------------------------------------------
template <int DINP, int DOUTP, int SIN, int SOUT, int ACT>  // ACT 0=tanh 1=sigmoid
__device__ __forceinline__ void wmma_layer(const float* actIn,
                                           const float* __restrict__ Wp,
                                           const float* __restrict__ bp,
                                           float* actOut, int lane) {
  const int m    = lane & 15;
  const int kofe = (lane >> 4);       // 0 | 1  (pair index offset)
  const int row0 = kofe << 3;         // 0 | 8
  const v2f* arow = (const v2f*)(actIn + m * SIN);
#pragma unroll
  for (int n0 = 0; n0 < DOUTP; n0 += 16) {
    const int col = n0 + m;
    const v2f* brow = (const v2f*)(Wp + col * DINP);
    v8f c = {};
#pragma unroll
    for (int kk = 0; kk < DINP; kk += 4) {
      v2f a = arow[(kk >> 1) + kofe];
      v2f b = brow[(kk >> 1) + kofe];
      c = __builtin_amdgcn_wmma_f32_16x16x4_f32(false, a, false, b,
                                                (short)0, c, false, false);
    }
    const float bv = bp[col];
#pragma unroll
    for (int v = 0; v < 8; ++v) {
      float x = c[v] + bv;
      x = (ACT == 0) ? TANHF(x) : 1.0f / (1.0f + __expf(-x));
      actOut[(row0 + v) * SOUT + col] = x;
    }
  }
}

// ---------------------------------------------------------------------------
// K0: pack weights/biases into zero-padded layouts in workspace (runs once per
// launch; ~85 KB). Bias jobs use din==1; src==nullptr emits zeros.
// ---------------------------------------------------------------------------
struct PackJob { const float* src; int dout, din, doutp, dinp, off; };
struct PackJobs { PackJob j[28]; };
__global__ void k_pack(PackJobs J, float* __restrict__ dstBase) {
  const unsigned tid  = blockIdx.x * blockDim.x + threadIdx.x;
  const unsigned nthr = gridDim.x * blockDim.x;
  for (int q = 0; q < 28; ++q) {
    const PackJob job = J.j[q];
    const int tot = job.doutp * job.dinp;
    for (int i = tid; i < (unsigned)tot; i += nthr) {
      const int r = i / job.dinp, cc = i % job.dinp;
      float v = 0.0f;
      if (job.src && r < job.dout && cc < job.din) v = job.src[r * job.din + cc];
      dstBase[job.off + i] = v;
    }
  }
}

// ---------------------------------------------------------------------------
// K1: recover src[]/dst[] from the one-hot incidence matrices in one
// non-temporal streaming pass (bandwidth-dominant: 1.07 GB ~= 46 us at
// 23.3 TB/s) and zero the per-node scatter accumulators.
// ---------------------------------------------------------------------------
__global__ void k_extract(const v4f* __restrict__ Ro4, const v4f* __restrict__ Ri4,
                          int* __restrict__ src, int* __restrict__ dst,
                          float* __restrict__ nodeO, float* __restrict__ nodeI) {
  const unsigned tid  = blockIdx.x * blockDim.x + threadIdx.x;
  const unsigned nthr = gridDim.x * blockDim.x;
  if (tid < (unsigned)N_NODES * 8) { nodeO[tid] = 0.0f; nodeI[tid] = 0.0f; }
  const unsigned NV4 = (unsigned)N_NODES * N_EDGES / 4u;
  for (unsigned i = tid; i < NV4; i += nthr) {
    const unsigned e = i * 4u;
    const int n = (int)(e >> 15);
    const int j = (int)(e & (N_EDGES - 1));
    v4f a = __builtin_nontemporal_load(&Ro4[i]);
    if (a.x != 0.0f) src[j]     = n;
    if (a.y != 0.0f) src[j + 1] = n;
    if (a.z != 0.0f) src[j + 2] = n;
    if (a.w != 0.0f) src[j + 3] = n;
    v4f b = __builtin_nontemporal_load(&Ri4[i]);
    if (b.x != 0.0f) dst[j]     = n;
    if (b.y != 0.0f) dst[j + 1] = n;
    if (b.z != 0.0f) dst[j + 2] = n;
    if (b.w != 0.0f) dst[j + 3] = n;
  }
}

struct K2P {
  const float *w10,*b10,*w11,*b11,*w12,*b12;
  const float *w20,*b20,*w21,*b21,*w22,*b22;
  const float *w30,*b30,*w31,*b31,*w32,*b32;
};
struct K3P { const float *w0,*b0,*w1,*b1,*w2,*b2,*w3,*b3,*w4,*b4; };

// ---------------------------------------------------------------------------
// K2: one wave per 16-edge tile: edge features E, n1/n2/n3 tanh MLPs via
// WMMA, emit H[32768,8], scatter-add w[j]*H[j] by src/dst (== Ro@eH, Ri@eH).
// ---------------------------------------------------------------------------
__global__ __launch_bounds__(32) void k_edge_mlp(
    const float* __restrict__ X, const int* __restrict__ src,
    const int* __restrict__ dst, const float* __restrict__ wE, K2P P,
    float* __restrict__ gH, float* __restrict__ nodeO, float* __restrict__ nodeI) {
  __shared__ float sE[16 * 8];
  __shared__ float sIn[16 * 88];
  __shared__ float sT[16 * 48];
  __shared__ float sU[16 * 48];
  __shared__ float sH1[16 * 48];
  __shared__ float sH2[16 * 48];
  const int lane = threadIdx.x;
  const int j0   = blockIdx.x * 16;

  // E = reshape_edges(X[src] - X[dst]), zero-padded to 8 cols
  for (int p = lane; p < 16 * 8; p += 32) {
    const int r = p >> 3, cc = p & 7;
    float v = 0.0f;
    if (cc < 6) {
      const int j = j0 + r;
      const float d = X[src[j] * 6 + cc] - X[dst[j] * 6 + cc];
      switch (cc) {
        case 0: case 1: v = d * (1.0f / 1300.0f); break;
        case 2:         v = d * (1.0f / 250.0f); break;
        case 3:         v = acosf(cosf(d)) * (1.0f / 1.57f); break;
        case 4:         v = fabsf(d) * 0.5f; break;
        default:        v = d * (-1.0f / 210.0f); break;
      }
    }
    sE[r * 8 + cc] = v;
  }
  __syncthreads();
  // n1: 6 -> 40 -> 40 -> 40
  wmma_layer< 8, 48,  8, 48, 0>(sE,  P.w10, P.b10, sT,  lane); __syncthreads();
  wmma_layer<40, 48, 48, 48, 0>(sT,  P.w11, P.b11, sU,  lane); __syncthreads();
  wmma_layer<40, 48, 48, 48, 0>(sU,  P.w12, P.b12, sH1, lane); __syncthreads();
  // n2 input = [H1 | E] (46 -> pad 48)
  for (int p = lane; p < 16 * 48; p += 32) {
    const int r = p / 48, cc = p % 48;
    float v = 0.0f;
    if (cc < 40)      v = sH1[r * 48 + cc];
    else if (cc < 46) v = sE[r * 8 + (cc - 40)];
    sIn[r * 48 + cc] = v;
  }
  __syncthreads();
  wmma_layer<48, 48, 48, 48, 0>(sIn, P.w20, P.b20, sT,  lane); __syncthreads();
  wmma_layer<40, 48, 48, 48, 0>(sT,  P.w21, P.b21, sU,  lane); __syncthreads();
  wmma_layer<40, 48, 48, 48, 0>(sU,  P.w22, P.b22, sH2, lane); __syncthreads();
  // n3 input = [H1 | H2 | E] (86 -> pad 88)
  for (int p = lane; p < 16 * 88; p += 32) {
    const int r = p / 88, cc = p % 88;
    float v = 0.0f;
    if (cc < 40)      v = sH1[r * 48 + cc];
    else if (cc < 80) v = sH2[r * 48 + (cc - 40)];
    else if (cc < 86) v = sE[r * 8 + (cc - 80)];
    sIn[r * 88 + cc] = v;
  }
  __syncthreads();
  wmma_layer<88, 48, 88, 48, 0>(sIn, P.w30, P.b30, sT, lane); __syncthreads();
  wmma_layer<40, 32, 48, 48, 0>(sT,  P.w31, P.b31, sU, lane); __syncthreads();
  wmma_layer<20, 16, 48, 48, 0>(sU,  P.w32, P.b32, sT, lane); __syncthreads();
  // H -> global; scatter w[j]*H[j] by src (nodeO) and dst (nodeI)
  for (int p = lane; p < 16 * 8; p += 32) {
    const int r = p >> 3, cc = p & 7;
    const int j = j0 + r;
    const float h = sT[r * 48 + cc];
    gH[j * 8 + cc] = h;
    const float v = wE[j] * h;
    atomicAdd(&nodeO[src[j] * 8 + cc], v);
    atomicAdd(&nodeI[dst[j] * 8 + cc], v);
  }
}

// ---------------------------------------------------------------------------
// K3: Ho[j]=nodeO[dst[j]], Hi[j]=nodeI[src[j]]; cls 24->40->40->20->10->1
// (tanh x4, sigmoid). NITER loop in the reference is idempotent -> once.
// ---------------------------------------------------------------------------
__global__ __launch_bounds__(32) void k_classifier(
    const int* __restrict__ src, const int* __restrict__ dst,
    const float* __restrict__ gH, const float* __restrict__ nodeO,
    const float* __restrict__ nodeI, K3P P, float* __restrict__ out) {
  __shared__ float sIn[16 * 24];
  __shared__ float sT[16 * 48];
  __shared__ float sU[16 * 48];
  const int lane = threadIdx.x;
  const int j0   = blockIdx.x * 16;
  for (int p = lane; p < 16 * 24; p += 32) {
    const int r = p / 24, cc = p % 24;
    const int j = j0 + r;
    float v;
    if (cc < 8)       v = nodeO[dst[j] * 8 + cc];
    else if (cc < 16) v = gH[j * 8 + (cc - 8)];
    else              v = nodeI[src[j] * 8 + (cc - 16)];
    sIn[r * 24 + cc] = v;
  }
  __syncthreads();
  wmma_layer<24, 48, 24, 48, 0>(sIn, P.w0, P.b0, sT, lane); __syncthreads();
  wmma_layer<40, 48, 48, 48, 0>(sT,  P.w1, P.b1, sU, lane); __syncthreads();
  wmma_layer<40, 32, 48, 48, 0>(sU,  P.w2, P.b2, sT, lane); __syncthreads();
  wmma_layer<20, 16, 48, 48, 0>(sT,  P.w3, P.b3, sU, lane); __syncthreads();
  wmma_layer<12, 16, 48, 48, 1>(sU,  P.w4, P.b4, sT, lane); __syncthreads();
  if (lane < 16) out[j0 + lane] = sT[lane * 48];
}

extern "C" void kernel_launch(void* const* d_in, const int* in_sizes, int n_in,
                              void* d_out, int out_size, void* d_ws, size_t ws_size,
                              hipStream_t stream) {
  (void)n_in; (void)out_size; (void)ws_size;
  const float *X, *Ro, *Ri, *wE;
  const float *n1W0,*n1W1,*n1b1,*n1W2,*n1b2;
  const float *n2W0,*n2W1,*n2b1,*n2W2,*n2b2;
  const float *n3W0,*n3W1,*n3b1,*n3W2;
  const float *cW0,*cb0,*cW1,*cb1,*cW2,*cb2,*cW3,*cb3,*cW4,*cb4;
  auto F = [&](int i) { return (const float*)d_in[i]; };
  if (in_sizes[0] == N_NODES * 6) {
    // dict insertion-order flattening: X, Ro, Ri, w, n1, n2, n3, cls
    X = F(0); Ro = F(1); Ri = F(2); wE = F(3);
    n1W0 = F(4);  n1W1 = F(5);  n1b1 = F(6);  n1W2 = F(7);  n1b2 = F(8);
    n2W0 = F(9);  n2W1 = F(10); n2b1 = F(11); n2W2 = F(12); n2b2 = F(13);
    n3W0 = F(14); n3W1 = F(15); n3b1 = F(16); n3W2 = F(17);
    cW0 = F(18); cb0 = F(19); cW1 = F(20); cb1 = F(21); cW2 = F(22);
    cb2 = F(23); cW3 = F(24); cb3 = F(25); cW4 = F(26); cb4 = F(27);
  } else {
    // jax pytree flattening (alphabetical dict keys): Ri, Ro, X, cls, n1, n2, n3, w
    Ri = F(0); Ro = F(1); X = F(2);
    cW0 = F(3); cb0 = F(4); cW1 = F(5); cb1 = F(6); cW2 = F(7);
    cb2 = F(8); cW3 = F(9); cb3 = F(10); cW4 = F(11); cb4 = F(12);
    n1W0 = F(13); n1W1 = F(14); n1b1 = F(15); n1W2 = F(16); n1b2 = F(17);
    n2W0 = F(18); n2W1 = F(19); n2b1 = F(20); n2W2 = F(21); n2b2 = F(22);
    n3W0 = F(23); n3W1 = F(24); n3b1 = F(25); n3W2 = F(26);
    wE = F(27);
  }
  // workspace carve-out
  int*   src   = (int*)d_ws;
  int*   dst   = src + N_EDGES;
  float* nodeO = (float*)(dst + N_EDGES);
  float* nodeI = nodeO + N_NODES * 8;
  float* gH    = nodeI + N_NODES * 8;
  float* packB = gH + N_EDGES * 8;

  // Build pack jobs (weights padded to [dout_pad x din_pad], biases to dout_pad;
  // nullptr src -> zeros). Order must match K2P/K3P fill below.
  PackJobs J; int nj = 0, off = 0;
  const float* packed[28];
  auto add = [&](const float* s, int dout, int din, int doutp, int dinp) {
    J.j[nj] = PackJob{s, dout, din, doutp, dinp, off};
    packed[nj] = packB + off;
    off += doutp * dinp; ++nj;
  };
  add(n1W0, 40,  6, 48,  8); add(nullptr, 0, 0, 48, 1);   // 0,1
  add(n1W1, 40, 40, 48, 40); add(n1b1, 40, 1, 48, 1);     // 2,3
  add(n1W2, 40, 40, 48, 40); add(n1b2, 40, 1, 48, 1);     // 4,5
  add(n2W0, 40, 46, 48, 48); add(nullptr, 0, 0, 48, 1);   // 6,7
  add(n2W1, 40, 40, 48, 40); add(n2b1, 40, 1, 48, 1);     // 8,9
  add(n2W2, 40, 40, 48, 40); add(n2b2, 40, 1, 48, 1);     // 10,11
  add(n3W0, 40, 86, 48, 88); add(nullptr, 0, 0, 48, 1);   // 12,13
  add(n3W1, 20, 40, 32, 40); add(n3b1, 20, 1, 32, 1);     // 14,15
  add(n3W2,  8, 20, 16, 20); add(nullptr, 0, 0, 16, 1);   // 16,17
  add(cW0,  40, 24, 48, 24); add(cb0, 40, 1, 48, 1);      // 18,19
  add(cW1,  40, 40, 48, 40); add(cb1, 40, 1, 48, 1);      // 20,21
  add(cW2,  20, 40, 32, 40); add(cb2, 20, 1, 32, 1);      // 22,23
  add(cW3,  10, 20, 16, 20); add(cb3, 10, 1, 16, 1);      // 24,25
  add(cW4,   1, 10, 16, 12); add(cb4,  1, 1, 16, 1);      // 26,27

  K2P p2 = {packed[0],  packed[1],  packed[2],  packed[3],  packed[4],  packed[5],
            packed[6],  packed[7],  packed[8],  packed[9],  packed[10], packed[11],
            packed[12], packed[13], packed[14], packed[15], packed[16], packed[17]};
  K3P p3 = {packed[18], packed[19], packed[20], packed[21], packed[22],
            packed[23], packed[24], packed[25], packed[26], packed[27]};

  k_pack<<<64, 256, 0, stream>>>(J, packB);
  k_extract<<<8192, 256, 0, stream>>>((const v4f*)Ro, (const v4f*)Ri, src, dst,
                                      nodeO, nodeI);
  k_edge_mlp<<<N_EDGES / 16, 32, 0, stream>>>(X, src, dst, wE, p2, gH, nodeO,
                                              nodeI);
  k_classifier<<<N_EDGES / 16, 32, 0, stream>>>(src, dst, gH, nodeO, nodeI, p3,
                                                (float*)d_out);
}